// FastSumOfParabolas_33998961115177
// MI455X (gfx1250) — compile-verified
//
#include <hip/hip_runtime.h>
#include <hip/hip_bf16.h>

// ---- constants from the reference ----
#define NUM_X    64
#define NUM_Y    64
#define NUM_BINS 128
#define NUM_SUB  10
#define NUM_V    (NUM_SUB * NUM_BINS)     // 1280
#define PAD      NUM_V                    // 1280
#define ROW_LEN  (NUM_V + 2 * PAD)        // 3840 floats per (y,x) row
#define BATCH    32
#define NY       64
#define NX       64

// V_RANGE = (c * numBins * t_res / 2)^2, computed in double like Python, used as f32
static constexpr double V_RANGE_D = (3.0e8 * (double)NUM_BINS * 32e-12 / 2.0) *
                                    (3.0e8 * (double)NUM_BINS * 32e-12 / 2.0); // 0.37748736
static constexpr float  V_BASE_F  = (float)(V_RANGE_D / (double)NUM_V);        // ~2.94912e-4
static constexpr float  STEP_F    = (float)(V_RANGE_D / (double)(NUM_BINS - 1)); // linspace step

typedef float floatx4 __attribute__((ext_vector_type(4)));

// One block per (b, y): 256 threads (8 wave32), 64 cells x 128 bins = 8192 outputs.
// Key structure: with g = it*256 + tid, (g & 31) is loop-invariant per thread, so each
// thread's four v-indices are computed ONCE into registers; only the 64-entry row-base
// table lives in LDS. The hot loop is 4 gathers + 1 NT b128 store + 1 prefetch, no
// intra-loop synchronization.
__global__ __launch_bounds__(256) void fsp_gather_kernel(
    const float* __restrict__ cur_pos,   // (B, NY, NX, 2)
    const float* __restrict__ deltas,    // (B, 3)
    const float* __restrict__ canon,     // (NUM_Y, NUM_X, ROW_LEN)
    const int*  __restrict__ is_cam,     // scalar
    float* __restrict__ out)             // (B, NY, NX, NUM_BINS)
{
    __shared__ int s_row[NX];          // row base (element offset) per x cell

    const int tid = threadIdx.x;
    const int blk = blockIdx.x;        // 0 .. B*NY-1
    const int b   = blk >> 6;          // / NY
    const int y   = blk & 63;          // % NY

    const float dxv  = deltas[b * 3 + 0];
    const float dyv  = deltas[b * 3 + 1];
    const float dzv  = deltas[b * 3 + 2];
    const float sign = (is_cam[0] != 0) ? 1.0f : -1.0f;

    // ---- per-thread v-indices (loop-invariant: bin group = tid & 31) ----
    // xlin = linspace(0, V_RANGE, NUM_BINS); v always subtracts the v-shift.
    // astype(int32) == C truncation toward zero; reference applies no clamp on v.
    const int bin0 = (tid & 31) << 2;
    int v0, v1, v2, v3;
    {
        float x0 = (float)(bin0 + 0) * STEP_F;
        float x1 = (float)(bin0 + 1) * STEP_F;
        float x2 = (float)(bin0 + 2) * STEP_F;
        float x3 = (float)(bin0 + 3) * STEP_F;
        v0 = PAD + (int)((-dzv + x0) / V_BASE_F);
        v1 = PAD + (int)((-dzv + x1) / V_BASE_F);
        v2 = PAD + (int)((-dzv + x2) / V_BASE_F);
        v3 = PAD + (int)((-dzv + x3) / V_BASE_F);
    }

    // ---- setup: per-cell row bases (threads 0..63) ----
    if (tid < NX) {
        const int x = tid;
        const float* cp = cur_pos + ((size_t)((b * NY + y) * NX + x)) * 2;
        float xs = fminf(fmaxf(cp[0] + sign * dxv, -1.0f), 1.0f);
        float ys = fminf(fmaxf(cp[1] + sign * dyv, -1.0f), 1.0f);
        // NUM * (s - MIN) / (MAX - MIN); divide-by-2 == *0.5 exactly in fp32
        int xi = (int)(((float)NUM_X * (xs + 1.0f)) * 0.5f);
        int yi = (int)(((float)NUM_Y * (ys + 1.0f)) * 0.5f);
        xi = min(NUM_X - 1, max(0, xi));
        yi = min(NUM_Y - 1, max(0, yi));
        s_row[x] = (yi * NUM_X + xi) * ROW_LEN;
    }
    __syncthreads();

    // ---- hot loop: 2048 float4 groups per block, 8 per thread ----
    // group g -> cell x = g>>5 = it*8 + (tid>>5); lanes of a wave cover one cell
    // contiguously -> coalesced B128 NT stores; gathers cluster in one ~5KB row span.
    float* out_base = out + ((size_t)(b * NY + y)) * (NX * NUM_BINS);
    const int xoff = tid >> 5;   // wave id within block

#pragma unroll
    for (int it = 0; it < 8; ++it) {
        const int g   = it * 256 + tid;      // 0..2047
        const int x   = it * 8 + xoff;       // == g >> 5
        const int row = s_row[x];

        // CDNA5 global_prefetch_b8: pull next iteration's row span toward the WGP
        if (it < 7) {
            __builtin_prefetch(&canon[(size_t)s_row[x + 8] + (size_t)v0], 0, 1);
        }

        floatx4 v;
        v.x = canon[(size_t)row + (size_t)v0];
        v.y = canon[(size_t)row + (size_t)v1];
        v.z = canon[(size_t)row + (size_t)v2];
        v.w = canon[(size_t)row + (size_t)v3];

        // Non-temporal: stream 64MB of output past L2 so the 63MB table stays resident
        __builtin_nontemporal_store(v, (floatx4*)(out_base + (size_t)g * 4));
    }
}

extern "C" void kernel_launch(void* const* d_in, const int* in_sizes, int n_in,
                              void* d_out, int out_size, void* d_ws, size_t ws_size,
                              hipStream_t stream) {
    (void)in_sizes; (void)n_in; (void)out_size; (void)d_ws; (void)ws_size;
    const float* cur_pos = (const float*)d_in[0];   // (32,64,64,2) f32
    const float* deltas  = (const float*)d_in[1];   // (32,3) f32
    const float* canon   = (const float*)d_in[2];   // (64,64,3840) f32
    const int*   is_cam  = (const int*)d_in[3];     // scalar int
    float*       out     = (float*)d_out;           // (32,64,64,128) f32

    dim3 grid(BATCH * NY);   // 2048 blocks
    dim3 block(256);         // 8 wave32 per block
    fsp_gather_kernel<<<grid, block, 0, stream>>>(cur_pos, deltas, canon, is_cam, out);
}